// BasisAffinityGAT_15805479649710
// MI455X (gfx1250) — compile-verified
//
#include <hip/hip_runtime.h>

// ---------------------------------------------------------------------------
// BasisAffinityGAT for MI455X (gfx1250, wave32, WMMA).
//
// Roofline: outputs are 2 * B*K*N*N f32 = 134 MB of stores -> ~6us at
// 23.3 TB/s. Total GEMM work ~2.4 GFLOP f32; with V_WMMA_F32_16X16X4_F32
// (~1/32 of the fp8 rate -> ~300 TFLOPS) that is ~7us IF throughput-bound,
// so the q/k projection is split 8-ways over the f dimension (16384 waves)
// to avoid being latency-bound, with deterministic partial-sum buffers.
//
// Math simplification: after symmetrization, logits[i,j] = 0.5*(s[i]+s[j])
// with s = e_q + e_k; the 0.5*s[i] term is constant along the softmax axis,
// so alpha[b,k,i,:] == softmax(0.5*s[b,k,:]) for every i.
// ---------------------------------------------------------------------------

typedef float v2f __attribute__((ext_vector_type(2)));
typedef float v8f __attribute__((ext_vector_type(8)));

#define B_ 8
#define N_ 512
#define D_ 128
#define K_ 8
#define R_ 128
#define MOM 0.99f
#define EPS_ 1e-6f
#define SLOPE 0.2f
#define FT_ 8            // f-tiles per (b,k) projection (R_/16)

__device__ __forceinline__ v8f wmma_f32(v2f a, v2f b, v8f c) {
  // D = A(16x4 f32) * B(4x16 f32) + C(16x16 f32)
  return __builtin_amdgcn_wmma_f32_16x16x4_f32(
      /*neg_a=*/false, a, /*neg_b=*/false, b,
      /*c_mod=*/(short)0, c, /*reuse_a=*/false, /*reuse_b=*/false);
}

// ---------------------------------------------------------------------------
// Kernel 1: fused = [desc | nv] @ W_fuse + b_fuse
// C is (B*N=4096) x 128, K-dim = 256 split between the two sources.
// One wave -> one 16x16 output tile; 64 WMMAs per tile (2048 waves).
// ---------------------------------------------------------------------------
__global__ void fuse_gemm_kernel(const float* __restrict__ desc,
                                 const float* __restrict__ nv,
                                 const float* __restrict__ Wf,
                                 const float* __restrict__ bf,
                                 float* __restrict__ fused) {
  const int wavesPerBlock = blockDim.x >> 5;
  const int waveId = blockIdx.x * wavesPerBlock + (threadIdx.x >> 5);
  const int lane = threadIdx.x & 31;
  const int half = lane >> 4;       // K-half for A/B fragments
  const int l    = lane & 15;       // row (A) / col (B) within fragment

  const int tilesN = D_ / 16;       // 8
  const int tM = waveId / tilesN;   // 0..255
  const int tN = waveId % tilesN;
  const int m0 = tM * 16, n0 = tN * 16;

  v8f acc = {};
  const int rowA = m0 + l;          // row in [0, B*N)

  const float* srcs[2] = {desc, nv};
  #pragma unroll
  for (int s = 0; s < 2; ++s) {
    const float* xrow = srcs[s] + (size_t)rowA * D_;
    for (int k0 = 0; k0 < D_; k0 += 4) {
      // A fragment: lanes 0-15 -> K=k0,k0+1 ; lanes 16-31 -> K=k0+2,k0+3
      v2f a; a.x = xrow[k0 + 2 * half]; a.y = xrow[k0 + 2 * half + 1];
      const int kr = s * D_ + k0 + 2 * half;
      v2f b; b.x = Wf[(size_t)kr * D_ + n0 + l];
             b.y = Wf[(size_t)(kr + 1) * D_ + n0 + l];
      acc = wmma_f32(a, b, acc);
    }
  }

  const int col = n0 + l;
  const float bias = bf[col];
  #pragma unroll
  for (int v = 0; v < 8; ++v) {
    const int row = m0 + v + 8 * half;   // C layout: VGPR v -> M=v / M=v+8
    fused[(size_t)row * D_ + col] = acc[v] + bias;
  }
}

// ---------------------------------------------------------------------------
// Kernel 2: split-f partial e computation.
// Wave (b,k,ft,nt) computes the 16x16 tile q[n0:n0+16, f0:f0+16] (and the
// matching k tile, sharing the A fragment), applies leaky-relu, weights by
// a_q/a_k, reduces over the 16 columns, and writes a deterministic partial:
//   eqp[ft][b*K+k][n] = sum_{f in tile} lrelu(q[n,f]) * a_q[f]
// 16384 waves -> high occupancy; no float atomics (bitwise reproducible).
// ---------------------------------------------------------------------------
__global__ void qk_e_partial_kernel(const float* __restrict__ fused,
                                    const float* __restrict__ Wq,
                                    const float* __restrict__ Wk,
                                    const float* __restrict__ avec,
                                    float* __restrict__ eqp,
                                    float* __restrict__ ekp) {
  const int wavesPerBlock = blockDim.x >> 5;
  const int waveId = blockIdx.x * wavesPerBlock + (threadIdx.x >> 5);
  const int lane = threadIdx.x & 31;
  const int half = lane >> 4;
  const int l    = lane & 15;

  // waveId = ((bk*FT_ + ft)*32 + nt): waves sharing a W tile are adjacent.
  const int nt = waveId & 31;
  const int ft = (waveId >> 5) & (FT_ - 1);
  const int bk = waveId >> 8;               // 0..63
  const int k  = bk & (K_ - 1);
  const int b  = bk >> 3;
  const int n0 = nt * 16, f0 = ft * 16;

  const float* frow = fused + (size_t)(b * N_ + n0 + l) * D_;
  const float* WqK  = Wq + (size_t)k * D_ * R_;
  const float* WkK  = Wk + (size_t)k * D_ * R_;

  v8f accq = {};
  v8f acck = {};
  for (int k0 = 0; k0 < D_; k0 += 4) {
    v2f a; a.x = frow[k0 + 2 * half]; a.y = frow[k0 + 2 * half + 1];
    const int kr = k0 + 2 * half;
    v2f bq; bq.x = WqK[(size_t)kr * R_ + f0 + l];
            bq.y = WqK[(size_t)(kr + 1) * R_ + f0 + l];
    v2f bkf; bkf.x = WkK[(size_t)kr * R_ + f0 + l];
             bkf.y = WkK[(size_t)(kr + 1) * R_ + f0 + l];
    accq = wmma_f32(a, bq, accq);
    acck = wmma_f32(a, bkf, acck);
  }

  const float aqv = avec[(size_t)k * 2 * R_ + f0 + l];        // a[k, f, 0]
  const float akv = avec[(size_t)k * 2 * R_ + R_ + f0 + l];   // a[k, R+f, 0]

  float epq[8], epk[8];
  #pragma unroll
  for (int v = 0; v < 8; ++v) {
    float x = accq[v]; x = (x > 0.f) ? x : SLOPE * x; epq[v] = x * aqv;
    float y = acck[v]; y = (y > 0.f) ? y : SLOPE * y; epk[v] = y * akv;
  }

  // Reduce over the 16 columns: C/D layout puts row v's columns across the
  // 16 lanes of each half -> xor-shuffle reduce within width-16 groups.
  #pragma unroll
  for (int v = 0; v < 8; ++v) {
    float xq = epq[v], xk = epk[v];
    for (int m = 8; m >= 1; m >>= 1) {
      xq += __shfl_xor(xq, m, 16);
      xk += __shfl_xor(xk, m, 16);
    }
    epq[v] = xq; epk[v] = xk;
  }

  if (l == 0) {   // lane 0 holds rows 0..7, lane 16 holds rows 8..15
    #pragma unroll
    for (int v = 0; v < 8; ++v) {
      const int n = n0 + v + 8 * half;
      const size_t o = ((size_t)ft * (B_ * K_) + bk) * N_ + n;
      eqp[o] = epq[v];
      ekp[o] = epk[v];
    }
  }
}

// ---------------------------------------------------------------------------
// Kernel 3: p[b,k,:] = softmax_j( 0.5 * (e_q[b,k,j] + e_k[b,k,j]) ),
// folding the 8 f-tile partials during the load. One block per (b,k).
// ---------------------------------------------------------------------------
__global__ void softmax_kernel(const float* __restrict__ eqp,
                               const float* __restrict__ ekp,
                               float* __restrict__ p) {
  const int bk = blockIdx.x;        // 0..B*K-1
  const int t = threadIdx.x;        // 512 threads
  __shared__ float red[N_];

  float s = 0.f;
  #pragma unroll
  for (int ft = 0; ft < FT_; ++ft) {
    const size_t o = ((size_t)ft * (B_ * K_) + bk) * N_ + t;
    s += eqp[o] + ekp[o];
  }
  const float x = 0.5f * s;

  red[t] = x;
  __syncthreads();
  for (int w = N_ / 2; w > 0; w >>= 1) {
    if (t < w) red[t] = fmaxf(red[t], red[t + w]);
    __syncthreads();
  }
  const float mx = red[0];
  __syncthreads();

  const float ex = __expf(x - mx);
  red[t] = ex;
  __syncthreads();
  for (int w = N_ / 2; w > 0; w >>= 1) {
    if (t < w) red[t] += red[t + w];
    __syncthreads();
  }
  p[(size_t)bk * N_ + t] = ex * (1.0f / red[0]);
}

// ---------------------------------------------------------------------------
// Kernel 4: pmean[k,n] = mean_b p[b,k,n]
// ---------------------------------------------------------------------------
__global__ void pmean_kernel(const float* __restrict__ p,
                             float* __restrict__ pmean) {
  const int i = blockIdx.x * blockDim.x + threadIdx.x;
  if (i >= K_ * N_) return;
  const int k = i / N_, n = i % N_;
  float s = 0.f;
  #pragma unroll
  for (int b = 0; b < B_; ++b) s += p[(size_t)(b * K_ + k) * N_ + n];
  pmean[i] = s * (1.0f / B_);
}

// ---------------------------------------------------------------------------
// Kernel 5: the bandwidth-bound writer (134 MB of stores, float4-vectorized).
//   alpha[b,k,i,j]    = p[b,k,j]                      (rows identical)
//   bias_log[b,k,i,j] = log(max(0.99*ema[k,i,j] + 0.01*pmean[k,j], eps))
// ema is 8 MB -> resident in the 192 MB L2 across the B=8 broadcast passes;
// a speculative global_prefetch_b8 pulls the next ema row ahead of use.
// ---------------------------------------------------------------------------
__global__ void write_out_kernel(const float* __restrict__ p,
                                 const float* __restrict__ pmean,
                                 const float* __restrict__ ema,
                                 float* __restrict__ out_bias,
                                 float* __restrict__ out_alpha) {
  const size_t tid = (size_t)blockIdx.x * blockDim.x + threadIdx.x;
  const int J4 = N_ / 4;
  const int j4 = (int)(tid % J4);
  size_t rest = tid / J4;
  const int i = (int)(rest % N_); rest /= N_;
  const int k = (int)(rest % K_);
  const int b = (int)(rest / K_);
  const int j0 = j4 * 4;

  const float* emap = ema + ((size_t)k * N_ + i) * N_ + j0;
  // Speculative prefetch of the next ema row (translation faults dropped).
  __builtin_prefetch(emap + N_, 0, 3);

  const float4 pm = *(const float4*)(pmean + (size_t)k * N_ + j0);
  const float4 pv = *(const float4*)(p + (size_t)(b * K_ + k) * N_ + j0);
  const float4 em = *(const float4*)emap;

  const float c1 = MOM, c2 = 1.0f - MOM;
  float4 bias;
  bias.x = __logf(fmaxf(c1 * em.x + c2 * pm.x, EPS_));
  bias.y = __logf(fmaxf(c1 * em.y + c2 * pm.y, EPS_));
  bias.z = __logf(fmaxf(c1 * em.z + c2 * pm.z, EPS_));
  bias.w = __logf(fmaxf(c1 * em.w + c2 * pm.w, EPS_));

  const size_t o = (((size_t)(b * K_ + k) * N_) + i) * N_ + j0;
  *(float4*)(out_bias + o)  = bias;
  *(float4*)(out_alpha + o) = pv;
}

// ---------------------------------------------------------------------------
extern "C" void kernel_launch(void* const* d_in, const int* in_sizes, int n_in,
                              void* d_out, int out_size, void* d_ws, size_t ws_size,
                              hipStream_t stream) {
  const float* desc = (const float*)d_in[0];   // [B,N,D]
  const float* nv   = (const float*)d_in[1];   // [B,N,D]
  const float* Wf   = (const float*)d_in[2];   // [2D,D]
  const float* bf   = (const float*)d_in[3];   // [D]
  const float* Wq   = (const float*)d_in[4];   // [K,D,R]
  const float* Wk   = (const float*)d_in[5];   // [K,D,R]
  const float* av   = (const float*)d_in[6];   // [K,2R,1]
  const float* ema  = (const float*)d_in[7];   // [K,N,N]

  float* ws = (float*)d_ws;
  float* fused = ws;                                   // B*N*D        = 524288
  float* eqp   = fused + (size_t)B_ * N_ * D_;         // FT_*B*K*N    = 262144
  float* ekp   = eqp + (size_t)FT_ * B_ * K_ * N_;     // FT_*B*K*N    = 262144
  float* p     = ekp + (size_t)FT_ * B_ * K_ * N_;     // B*K*N        =  32768
  float* pmean = p + (size_t)B_ * K_ * N_;             // K*N          =   4096

  float* out_bias  = (float*)d_out;
  float* out_alpha = out_bias + (size_t)B_ * K_ * N_ * N_;

  // 1) fuse GEMM: 2048 16x16 tiles, 4 waves/block
  fuse_gemm_kernel<<<512, 128, 0, stream>>>(desc, nv, Wf, bf, fused);

  // 2) q/k projections, split over f: B*K*FT*(N/16) = 16384 waves
  qk_e_partial_kernel<<<4096, 128, 0, stream>>>(fused, Wq, Wk, av, eqp, ekp);

  // 3) B*K row softmaxes of length N (folds the 8 partials)
  softmax_kernel<<<B_ * K_, N_, 0, stream>>>(eqp, ekp, p);

  // 4) mean over batch
  pmean_kernel<<<(K_ * N_ + 255) / 256, 256, 0, stream>>>(p, pmean);

  // 5) 134 MB writer: B*K*N*(N/4) threads
  const size_t total = (size_t)B_ * K_ * N_ * (N_ / 4);
  write_out_kernel<<<(unsigned)(total / 256), 256, 0, stream>>>(
      p, pmean, ema, out_bias, out_alpha);
}